// LocalMoranIndex_11244224381607
// MI455X (gfx1250) — compile-verified
//
#include <hip/hip_runtime.h>

// ---------------------------------------------------------------------------
// Local Moran's I on gfx1250 (MI455X).
//  - mean(X): deterministic two-stage reduction (no float atomics)
//  - main:    1 wave per 16 rows; K=32 reduction done with
//             V_WMMA_F32_16X16X4_F32 row-sum trick (B = ones), two
//             accumulator chains (lag, sum w*xn^2), full fp32 precision.
//  - W / ids streamed with non-temporal loads so the 4 MB X array stays
//    resident in the 192 MB L2 for the 32M random gathers.
// ---------------------------------------------------------------------------

typedef float v2f __attribute__((ext_vector_type(2)));
typedef int   v2i __attribute__((ext_vector_type(2)));
typedef float v8f __attribute__((ext_vector_type(8)));

#define RED_BLOCKS 256
#define RED_THREADS 256
#define KNB 32              // neighbors per row (reference: K = 32)

__global__ __launch_bounds__(RED_THREADS)
void lmi_sum_partial(const float* __restrict__ X, float* __restrict__ partial, int n) {
    __shared__ float smem[RED_THREADS];
    float acc = 0.0f;
    for (int i = blockIdx.x * RED_THREADS + threadIdx.x; i < n; i += gridDim.x * RED_THREADS)
        acc += X[i];
    smem[threadIdx.x] = acc;
    __syncthreads();
    #pragma unroll
    for (int s = RED_THREADS / 2; s > 0; s >>= 1) {
        if ((int)threadIdx.x < s) smem[threadIdx.x] += smem[threadIdx.x + s];
        __syncthreads();
    }
    if (threadIdx.x == 0) partial[blockIdx.x] = smem[0];
}

__global__ __launch_bounds__(RED_BLOCKS)
void lmi_finalize_mean(const float* __restrict__ partial, float* __restrict__ mean_out, int n) {
    __shared__ float smem[RED_BLOCKS];
    smem[threadIdx.x] = partial[threadIdx.x];
    __syncthreads();
    #pragma unroll
    for (int s = RED_BLOCKS / 2; s > 0; s >>= 1) {
        if ((int)threadIdx.x < s) smem[threadIdx.x] += smem[threadIdx.x + s];
        __syncthreads();
    }
    if (threadIdx.x == 0) mean_out[0] = smem[0] / (float)n;
}

// One wave32 handles a 16-row tile. A-matrix 16x4 f32 layout (ISA 7.12.2):
//   lanes 0-15 / 16-31 both map to M = lane%16; VGPR0/VGPR1 x lane-half give
//   the 4 K slots. B = all ones  =>  D[m][*] = sum over the 4 A values of row m,
//   accumulated over 8 chunks => full K=32 row sum, exact fp32.
__global__ __launch_bounds__(256)
void lmi_main(const float* __restrict__ X,
              const float* __restrict__ W,
              const int*   __restrict__ IDS,
              const float* __restrict__ meanp,
              float*       __restrict__ out,
              int n) {
    const int lane = threadIdx.x & 31;
    const int wave = threadIdx.x >> 5;
    const int tile = blockIdx.x * 8 + wave;          // wave-uniform
    const int full_tiles = n >> 4;
    const float mean = meanp[0];

    if (tile < full_tiles) {
        const int base = tile << 4;                  // first row of tile
        const int m    = lane & 15;                  // row within tile (M)
        const int h    = lane >> 4;                  // lane half -> K slots
        const long long rowoff = (long long)(base + m) * KNB;
        const float* wrow = W   + rowoff;
        const int*   irow = IDS + rowoff;

        v8f dlag = {0.f,0.f,0.f,0.f,0.f,0.f,0.f,0.f};
        v8f dsq  = {0.f,0.f,0.f,0.f,0.f,0.f,0.f,0.f};
        v2f ones; ones.x = 1.0f; ones.y = 1.0f;      // B (4x16) = all ones

        #pragma unroll
        for (int c = 0; c < 8; ++c) {
            const int k = c * 4 + h * 2;             // adjacent pair -> b64 loads
            v2f wp = __builtin_nontemporal_load(
                         reinterpret_cast<const v2f*>(wrow + k));
            v2i ip = __builtin_nontemporal_load(
                         reinterpret_cast<const v2i*>(irow + k));
            // gathers from X: regular (temporal) loads -> X stays hot in L2
            const float xn0 = X[ip.x] - mean;
            const float xn1 = X[ip.y] - mean;
            v2f A;  A.x  = wp.x * xn0;       A.y  = wp.y * xn1;        // w*xn
            v2f Aq; Aq.x = A.x  * xn0;       Aq.y = A.y  * xn1;        // w*xn^2
            dlag = __builtin_amdgcn_wmma_f32_16x16x4_f32(
                       false, A,  false, ones, (short)0, dlag, false, false);
            dsq  = __builtin_amdgcn_wmma_f32_16x16x4_f32(
                       false, Aq, false, ones, (short)0, dsq,  false, false);
        }

        // D layout: lane 0 holds rows 0-7 (VGPR v -> M=v), lane 16 rows 8-15.
        if ((lane & 15) == 0) {
            const int rbase = base + h * 8;
            #pragma unroll
            for (int v = 0; v < 8; ++v) {
                const float xs  = X[rbase + v] - mean;
                const float lag = dlag[v];
                const float sq  = dsq[v];
                __builtin_nontemporal_store(xs * lag * (float)(KNB - 1) / sq,
                                            out + rbase + v);
            }
        }
    } else if ((n & 15) && tile == full_tiles) {
        // scalar tail (unused for N = 1e6, kept for generality)
        const int rem = n & 15;
        if (lane < rem) {
            const int row = full_tiles * 16 + lane;
            const float* wrow = W   + (long long)row * KNB;
            const int*   irow = IDS + (long long)row * KNB;
            float lag = 0.f, sq = 0.f;
            for (int k = 0; k < KNB; ++k) {
                const float xn = X[irow[k]] - mean;
                const float p  = wrow[k] * xn;
                lag += p; sq += p * xn;
            }
            const float xs = X[row] - mean;
            out[row] = xs * lag * (float)(KNB - 1) / sq;
        }
    }
}

extern "C" void kernel_launch(void* const* d_in, const int* in_sizes, int n_in,
                              void* d_out, int out_size, void* d_ws, size_t ws_size,
                              hipStream_t stream) {
    const float* X   = (const float*)d_in[0];
    const float* W   = (const float*)d_in[1];
    const int*   IDS = (const int*)d_in[2];
    float*       out = (float*)d_out;
    const int n = in_sizes[0];

    float* partial = (float*)d_ws;           // RED_BLOCKS floats
    float* meanp   = partial + RED_BLOCKS;   // 1 float

    lmi_sum_partial<<<RED_BLOCKS, RED_THREADS, 0, stream>>>(X, partial, n);
    lmi_finalize_mean<<<1, RED_BLOCKS, 0, stream>>>(partial, meanp, n);

    const int tiles  = (n + 15) / 16;        // includes tail wave if any
    const int blocks = (tiles + 7) / 8;      // 8 waves (256 thr) per block
    lmi_main<<<blocks, 256, 0, stream>>>(X, W, IDS, meanp, out, n);
}